// TensorProductScoreModel_71536975282766
// MI455X (gfx1250) — compile-verified
//
#include <hip/hip_runtime.h>
#include <hip/hip_bf16.h>
#include <math.h>

// ---------------------------------------------------------------------------
// TensorProductScoreModel fused edge kernel for gfx1250 (MI455X, wave32).
//
//   hidden = relu(edge_attr @ fc1_w + fc1_b)          [E,48]
//   w      = hidden @ fc2_w + fc2_b                   [E,320]  (never hits HBM)
//   out0   = einsum('euw,eu->ew', w0, x) * 0.25       [E,16]
//   out1   = einsum('euv,eu->ev', w1, x) (x) sh1 *.25 [E,4,3]
//   segment-mean over edge_src                        [N,28]
//
// 8 waves / block share one LDS copy of the weights stored in a
// FRAGMENT-READY layout: each lane's B fragment (2 consecutive K values of
// one column) is one aligned ds_load_b64, and a wave's fragment read is 256
// contiguous bytes -> 64 banks, conflict-free, zero packing moves.
// All matmuls via V_WMMA_F32_16X16X4_F32 (fp32-exact vs reference).
// ---------------------------------------------------------------------------

typedef float v2f __attribute__((ext_vector_type(2)));
typedef float v8f __attribute__((ext_vector_type(8)));

#define NSF    16
#define HID    48
#define WNUM   320
#define MSGW   28
#define HSTR   50      // hidden-tile LDS stride (conflict-free, 8B aligned)
#define NKC    12      // K chunks of 4 (K = 48)
#define NWAVES 8
// fragment-ready sizes: [nt][kc][half][n][2]
#define W1F_SZ (3 * NKC * 2 * NSF * 2)    // 2304 floats
#define W2F_SZ (20 * NKC * 2 * NSF * 2)   // 15360 floats

__device__ __forceinline__ v8f wmma_f32_k4(v2f a, v2f b, v8f c) {
    // D = A(16x4 f32) * B(4x16 f32) + C(16x16 f32)
    return __builtin_amdgcn_wmma_f32_16x16x4_f32(
        /*neg_a=*/false, a, /*neg_b=*/false, b,
        /*c_mod=*/(short)0, c, /*reuse_a=*/false, /*reuse_b=*/false);
}

// wave-local ordering for wave-private LDS scratch (no block barrier needed)
__device__ __forceinline__ void wave_lds_sync() {
    __builtin_amdgcn_wave_barrier();
    asm volatile("s_wait_dscnt 0x0" ::: "memory");
}

__global__ __launch_bounds__(32 * NWAVES)
void tps_edge_kernel(const float* __restrict__ node_attr,   // [N,16]
                     const float* __restrict__ edge_attr,   // [E,48]
                     const float* __restrict__ edge_vec,    // [E,3]
                     const float* __restrict__ fc1_w,       // [48,48]
                     const float* __restrict__ fc1_b,       // [48]
                     const float* __restrict__ fc2_w,       // [48,320]
                     const float* __restrict__ fc2_b,       // [320]
                     const int*   __restrict__ edge_index,  // [2,E]
                     float* __restrict__ sums,              // [N,28]
                     float* __restrict__ cnts,              // [N]
                     int E, int numTiles)
{
    // fragment-ready shared weights, staged once per block
    __shared__ __align__(16) float w1f[W1F_SZ];
    __shared__ __align__(16) float w2f[W2F_SZ];
    __shared__ float b1s[HID];
    __shared__ float b2s[WNUM];
    // per-wave scratch
    __shared__ __align__(16) float hbuf [NWAVES][NSF * HSTR];
    __shared__ float xbuf [NWAVES][NSF * NSF];
    __shared__ float o1buf[NWAVES][NSF * 4];
    __shared__ float shbuf[NWAVES][NSF * 3];
    __shared__ int   srcbuf[NWAVES][NSF];

    const int tid  = threadIdx.x;
    const int wid  = tid >> 5;
    const int lane = tid & 31;
    const int half = lane >> 4;
    const int nn   = lane & 15;

    // ---- stage weights into LDS in fragment-ready layout ------------------
    // idx = (((nt*12 + kc)*2 + half)*16 + n)*2 + t
    //   -> row = kc*4 + half*2 + t ; col = nt*16 + n
    for (int i = tid; i < W1F_SZ; i += 32 * NWAVES) {
        const int t  = i & 1;
        const int n  = (i >> 1) & 15;
        const int hf = (i >> 5) & 1;
        const int kc = (i >> 6) % NKC;
        const int nt = i / (NKC * 64);
        w1f[i] = fc1_w[(kc * 4 + hf * 2 + t) * HID + nt * NSF + n];
    }
    for (int i = tid; i < W2F_SZ; i += 32 * NWAVES) {
        const int t  = i & 1;
        const int n  = (i >> 1) & 15;
        const int hf = (i >> 5) & 1;
        const int kc = (i >> 6) % NKC;
        const int nt = i / (NKC * 64);
        w2f[i] = fc2_w[(kc * 4 + hf * 2 + t) * WNUM + nt * NSF + n];
    }
    for (int i = tid; i < HID;  i += 32 * NWAVES) b1s[i] = fc1_b[i];
    for (int i = tid; i < WNUM; i += 32 * NWAVES) b2s[i] = fc2_b[i];
    __syncthreads();

    float* hb  = hbuf[wid];
    float* xb  = xbuf[wid];
    float* o1b = o1buf[wid];
    float* shb = shbuf[wid];
    int*   srb = srcbuf[wid];

    const int waveStride = gridDim.x * NWAVES;

    for (int tile = blockIdx.x * NWAVES + wid; tile < numTiles; tile += waveStride) {
        const int e0 = tile * NSF;

        // prefetch next tile's edge_attr row for this lane's edge
        {
            const int tn = tile + waveStride;
            if (tn < numTiles)
                __builtin_prefetch(edge_attr + (size_t)(tn * NSF + nn) * HID, 0, 1);
        }

        // ---- per-edge metadata: lanes 0..15 own edge m = lane -------------
        for (int i = lane; i < NSF * 4; i += 32) o1b[i] = 0.0f;
        if (lane < NSF) {
            const int e  = e0 + lane;
            const int ec = (e < E) ? e : (E - 1);
            srb[lane] = edge_index[ec];
            const int dst = edge_index[E + ec];
            #pragma unroll
            for (int u = 0; u < NSF; ++u)
                xb[lane * NSF + u] = node_attr[(size_t)dst * NSF + u];
            const float vx = edge_vec[(size_t)ec * 3 + 0];
            const float vy = edge_vec[(size_t)ec * 3 + 1];
            const float vz = edge_vec[(size_t)ec * 3 + 2];
            const float rn = sqrtf(vx * vx + vy * vy + vz * vz) + 1e-8f;
            const float c1 = 1.7320508075688772f;   // sqrt(3)
            shb[lane * 3 + 0] = c1 * vx / rn;
            shb[lane * 3 + 1] = c1 * vy / rn;
            shb[lane * 3 + 2] = c1 * vz / rn;
        }
        wave_lds_sync();

        // ---- A fragments: edge_attr tile [16 x 48] ------------------------
        // A 16x4 f32: lanes 0-15 M=0..15 K={0,1}; lanes 16-31 M=0..15 K={2,3}
        const int em = (e0 + nn < E) ? (e0 + nn) : (E - 1);
        const float* arow = edge_attr + (size_t)em * HID;
        v2f afrag[NKC];
        #pragma unroll
        for (int kc = 0; kc < NKC; ++kc)
            afrag[kc] = *(const v2f*)(arow + kc * 4 + half * 2);

        // per-lane fragment offset inside one (nt) slab of the weight arrays
        const int fo = (half * NSF + nn) * 2;     // bytes/4 within 64-dword slab

        // ---- GEMM1: hidden = relu(A @ W1 + b1) ----------------------------
        #pragma unroll
        for (int nt = 0; nt < 3; ++nt) {
            const float* wslab = w1f + nt * (NKC * 64);
            v2f bf[NKC];
            #pragma unroll
            for (int kc = 0; kc < NKC; ++kc)
                bf[kc] = *(const v2f*)(wslab + kc * 64 + fo);   // ds_load_b64
            v8f acc = {};
            #pragma unroll
            for (int kc = 0; kc < NKC; ++kc)
                acc = wmma_f32_k4(afrag[kc], bf[kc], acc);
            const float bv = b1s[nt * NSF + nn];
            // D layout: acc[j] = element (row j + 8*half, col nt*16+nn)
            #pragma unroll
            for (int j = 0; j < 8; ++j) {
                float v = acc[j] + bv;
                v = v > 0.0f ? v : 0.0f;
                hb[(j + half * 8) * HSTR + nt * NSF + nn] = v;
            }
        }
        wave_lds_sync();

        // ---- A fragments for GEMM2 from hidden tile (aligned b64 loads) ---
        v2f hfrag[NKC];
        #pragma unroll
        for (int kc = 0; kc < NKC; ++kc)
            hfrag[kc] = *(const v2f*)(hb + nn * HSTR + kc * 4 + half * 2);

        // ---- GEMM2 + fused contraction with x -----------------------------
        float acc0[8];
        #pragma unroll
        for (int j = 0; j < 8; ++j) acc0[j] = 0.0f;

        #pragma unroll
        for (int nt = 0; nt < 20; ++nt) {
            const float* wslab = w2f + nt * (NKC * 64);
            v2f bf[NKC];
            #pragma unroll
            for (int kc = 0; kc < NKC; ++kc)
                bf[kc] = *(const v2f*)(wslab + kc * 64 + fo);   // ds_load_b64
            v8f acc = {};
            #pragma unroll
            for (int kc = 0; kc < NKC; ++kc)
                acc = wmma_f32_k4(hfrag[kc], bf[kc], acc);
            const float bv = b2s[nt * NSF + nn];
            if (nt < 16) {
                // w0 slice u = nt: out0[m][nn] += w0[m][nt][nn] * x[m][nt]
                #pragma unroll
                for (int j = 0; j < 8; ++j) {
                    const int m = j + half * 8;
                    acc0[j] += (acc[j] + bv) * xb[m * NSF + nt];
                }
            } else {
                // w1 cols: u = (nt-16)*4 + nn/4, v = nn&3
                const int u = (nt - 16) * 4 + (nn >> 2);
                const int v = nn & 3;
                #pragma unroll
                for (int j = 0; j < 8; ++j) {
                    const int m = j + half * 8;
                    atomicAdd(&o1b[m * 4 + v], (acc[j] + bv) * xb[m * NSF + u]);
                }
            }
        }
        wave_lds_sync();

        // ---- emit messages into global segment accumulators ---------------
        const float inv = 0.25f;   // 1/sqrt(16); sh0 == 1
        #pragma unroll
        for (int j = 0; j < 8; ++j) {
            const int m = j + half * 8;
            if (e0 + m < E)
                atomicAdd(&sums[(size_t)srb[m] * MSGW + nn], acc0[j] * inv);
        }
        if (lane < NSF && (e0 + lane) < E)
            atomicAdd(&cnts[srb[lane]], 1.0f);
        for (int idx = lane; idx < NSF * 12; idx += 32) {
            const int m = idx / 12;
            const int c = idx % 12;        // c = v*3 + comp
            if (e0 + m < E) {
                const float val = o1b[m * 4 + (c / 3)] * shb[m * 3 + (c % 3)] * inv;
                atomicAdd(&sums[(size_t)srb[m] * MSGW + NSF + c], val);
            }
        }
        wave_lds_sync();   // o1b reused next tile
    }
}

__global__ void tps_zero_kernel(float* __restrict__ p, int n) {
    for (int i = blockIdx.x * blockDim.x + threadIdx.x; i < n;
         i += gridDim.x * blockDim.x)
        p[i] = 0.0f;
}

__global__ void tps_finalize_kernel(const float* __restrict__ sums,
                                    const float* __restrict__ cnts,
                                    float* __restrict__ out, int total) {
    const int i = blockIdx.x * blockDim.x + threadIdx.x;
    if (i < total) {
        const float c = cnts[i / MSGW];
        out[i] = sums[i] / fmaxf(c, 1.0f);
    }
}

extern "C" void kernel_launch(void* const* d_in, const int* in_sizes, int n_in,
                              void* d_out, int out_size, void* d_ws, size_t ws_size,
                              hipStream_t stream) {
    const float* node_attr  = (const float*)d_in[0];
    const float* edge_attr  = (const float*)d_in[1];
    const float* edge_vec   = (const float*)d_in[2];
    const float* fc1_w      = (const float*)d_in[3];
    const float* fc1_b      = (const float*)d_in[4];
    const float* fc2_w      = (const float*)d_in[5];
    const float* fc2_b      = (const float*)d_in[6];
    const int*   edge_index = (const int*)d_in[7];

    const int N = in_sizes[0] / NSF;
    const int E = in_sizes[2] / 3;

    float* sums = (float*)d_ws;                    // [N*28]
    float* cnts = sums + (size_t)N * MSGW;         // [N]

    const int ztot = N * (MSGW + 1);
    tps_zero_kernel<<<(ztot + 255) / 256, 256, 0, stream>>>(sums, ztot);

    const int tiles = (E + NSF - 1) / NSF;
    int blocks = (tiles + NWAVES - 1) / NWAVES;
    if (blocks > 1024) blocks = 1024;              // grid-stride, amortize staging
    tps_edge_kernel<<<blocks, 32 * NWAVES, 0, stream>>>(
        node_attr, edge_attr, edge_vec, fc1_w, fc1_b, fc2_w, fc2_b,
        edge_index, sums, cnts, E, tiles);

    const int total = N * MSGW;
    tps_finalize_kernel<<<(total + 255) / 256, 256, 0, stream>>>(
        sums, cnts, (float*)d_out, total);
}